// ReparamEmbeddings_88570815578781
// MI455X (gfx1250) — compile-verified
//
#include <hip/hip_runtime.h>
#include <hip/hip_bf16.h>

#define D_DIM 1024
#define R_DIM 16
#define WG    256

#if __has_builtin(__builtin_amdgcn_global_load_async_to_lds_b128) && __has_builtin(__builtin_amdgcn_s_wait_asynccnt)
#define HAVE_ASYNC 1
typedef int v4i __attribute__((vector_size(16)));
typedef __attribute__((address_space(1))) v4i gv4i_t;   // global AS operand
typedef __attribute__((address_space(3))) v4i lv4i_t;   // LDS AS operand
#endif

#if __has_builtin(__builtin_amdgcn_wmma_f32_16x16x4_f32)
#define HAVE_WMMA 1
#endif

typedef __attribute__((ext_vector_type(2))) float v2f;
typedef __attribute__((ext_vector_type(8))) float v8f;

__global__ __launch_bounds__(WG) void reparam_embed_kernel(
    const int* __restrict__ indices, const int* __restrict__ index_map,
    const float* __restrict__ emb, const float* __restrict__ down,
    const float* __restrict__ up, float* __restrict__ out)
{
    __shared__ float s_up[R_DIM * D_DIM];      // 64 KB up[idx] tile (async DMA target)
    __shared__ float s_hpart[(WG / 32) * R_DIM];
    __shared__ float s_h[R_DIM];

    const int n    = blockIdx.x;
    const int tid  = threadIdx.x;
    const int lane = tid & 31;
    const int wave = tid >> 5;

    const int idx  = indices[n];
    const int orig = index_map[idx];

    const float* e_row   = emb  + (size_t)orig * D_DIM;
    const float* down_m  = down + (size_t)idx * (size_t)(D_DIM * R_DIM);
    const float* up_m    = up   + (size_t)idx * (size_t)(R_DIM * D_DIM);
    float*       out_row = out  + (size_t)n * D_DIM;

    // ---- Phase A: async-copy up[idx] (64KB) into LDS; overlaps with Phase B ----
#if defined(HAVE_ASYNC)
    #pragma unroll
    for (int i = 0; i < (R_DIM * D_DIM) / (WG * 4); ++i) {   // 16 x b128 per thread
        const int off = (i * WG + tid) * 4;                  // float offset, 16B granules
        __builtin_amdgcn_global_load_async_to_lds_b128(
            (gv4i_t*)(const void*)(up_m + off),
            (lv4i_t*)(void*)(s_up + off), 0, 0);
    }
#else
    for (int i = 0; i < (R_DIM * D_DIM) / (WG * 4); ++i) {
        const int off = (i * WG + tid) * 4;
        *(float4*)(s_up + off) = *(const float4*)(up_m + off);
    }
#endif

    // ---- Phase B: h[r] = sum_d e[d] * down[d][r]  (streams down via b128 loads) ----
    float h[R_DIM];
    #pragma unroll
    for (int r = 0; r < R_DIM; ++r) h[r] = 0.0f;

    #pragma unroll
    for (int i = 0; i < D_DIM / WG; ++i) {
        const int d = i * WG + tid;
        const float ev = e_row[d];
        const float4* drow = (const float4*)(down_m + (size_t)d * R_DIM);
        const float4 c0 = drow[0], c1 = drow[1], c2 = drow[2], c3 = drow[3];
        h[0]  += ev * c0.x; h[1]  += ev * c0.y; h[2]  += ev * c0.z; h[3]  += ev * c0.w;
        h[4]  += ev * c1.x; h[5]  += ev * c1.y; h[6]  += ev * c1.z; h[7]  += ev * c1.w;
        h[8]  += ev * c2.x; h[9]  += ev * c2.y; h[10] += ev * c2.z; h[11] += ev * c2.w;
        h[12] += ev * c3.x; h[13] += ev * c3.y; h[14] += ev * c3.z; h[15] += ev * c3.w;
    }

    // wave32 butterfly reduction, then cross-wave combine in LDS
    #pragma unroll
    for (int r = 0; r < R_DIM; ++r) {
        float v = h[r];
        v += __shfl_xor(v, 16, 32);
        v += __shfl_xor(v,  8, 32);
        v += __shfl_xor(v,  4, 32);
        v += __shfl_xor(v,  2, 32);
        v += __shfl_xor(v,  1, 32);
        h[r] = v;
    }
    if (lane == 0) {
        #pragma unroll
        for (int r = 0; r < R_DIM; ++r) s_hpart[wave * R_DIM + r] = h[r];
    }
    __syncthreads();
    if (tid < R_DIM) {
        float s = 0.0f;
        #pragma unroll
        for (int w = 0; w < WG / 32; ++w) s += s_hpart[w * R_DIM + tid];
        s_h[tid] = s;
    }
#if defined(HAVE_ASYNC)
    __builtin_amdgcn_s_wait_asynccnt(0);   // my async tile writes are complete
#endif
    __syncthreads();                        // everyone's writes + s_h visible

    // ---- Phase C: out[d] = sum_r h[r] * up[r][d] ----
#if defined(HAVE_WMMA)
    {
        // A-matrix (16x4 f32): lanes 0-15 hold K=0 (v0), K=1 (v1); lanes 16-31 K=2,K=3.
        // Broadcasting h over all 16 M-rows makes every C row equal h . U_tile.
        const int koff = (lane >> 4) << 1;     // 0 or 2
        const int col  = lane & 15;
        #pragma unroll
        for (int t = 0; t < (D_DIM / (WG / 32)) / 16; ++t) {   // 8 tiles of 16 per wave
            const int base = wave * (D_DIM / (WG / 32)) + t * 16;
            v8f acc = {0.f, 0.f, 0.f, 0.f, 0.f, 0.f, 0.f, 0.f};
            #pragma unroll
            for (int c = 0; c < R_DIM / 4; ++c) {              // K = 16 via 4 x (K=4)
                v2f a, b;
                a.x = s_h[4 * c + koff];
                a.y = s_h[4 * c + koff + 1];
                b.x = s_up[(4 * c + koff)     * D_DIM + base + col];
                b.y = s_up[(4 * c + koff + 1) * D_DIM + base + col];
                acc = __builtin_amdgcn_wmma_f32_16x16x4_f32(
                          false, a, false, b, (short)0, acc, false, false);
            }
            if (lane < 16) out_row[base + col] = acc[0];       // row 0 of C
        }
    }
#else
    for (int i = 0; i < D_DIM / WG; ++i) {
        const int d = i * WG + tid;
        float s = 0.0f;
        #pragma unroll
        for (int r = 0; r < R_DIM; ++r) s += s_h[r] * s_up[r * D_DIM + d];
        out_row[d] = s;
    }
#endif
}

extern "C" void kernel_launch(void* const* d_in, const int* in_sizes, int n_in,
                              void* d_out, int out_size, void* d_ws, size_t ws_size,
                              hipStream_t stream) {
    const int*   indices   = (const int*)d_in[0];
    const int*   index_map = (const int*)d_in[1];
    const float* emb       = (const float*)d_in[2];
    const float* down      = (const float*)d_in[3];
    const float* up        = (const float*)d_in[4];
    float*       out       = (float*)d_out;

    const int n = in_sizes[0];   // N = 16384 indices, one block per index
    reparam_embed_kernel<<<n, WG, 0, stream>>>(indices, index_map, emb, down, up, out);
}